// LightweightMultiHeadAttention_5514738008777
// MI455X (gfx1250) — compile-verified
//
#include <hip/hip_runtime.h>
#include <hip/hip_bf16.h>

// ---------------------------------------------------------------------------
// LightweightMultiHeadAttention, fused single-pass kernel for gfx1250.
//   B=16384, S=60, D_IN=256, H=16, dk=8, dph=16
//   q_eff[h,:] = Q[h,:] @ Wk[h*8:(h+1)*8, :]   (folds query into key proj)
//   scores[s,h] = (v[b,s,:]·q_eff[h,:] + b_eff[h]) / sqrt(8)
//   attn = softmax_s(scores);  out[b,h,d] = sum_s attn * v[b,s,h*16+d]
// One workgroup per batch b; v[b] staged once in LDS (TDM if available).
// Score GEMM [64,256]x[256,16] via V_WMMA_F32_16X16X4_F32 (4 waves x 64 wmma).
// ---------------------------------------------------------------------------

typedef __attribute__((ext_vector_type(2)))  float        v2f;
typedef __attribute__((ext_vector_type(8)))  float        v8f;
typedef __attribute__((ext_vector_type(4)))  unsigned int v4u;
typedef __attribute__((ext_vector_type(4)))  int          v4i;
typedef __attribute__((ext_vector_type(8)))  int          v8i;

#define H_     16
#define DK_    8
#define DIN_   256
#define S_     60
#define SROW   64           // S padded to 64 rows for 4 WMMA row-tiles
#define SV_STRIDE 260       // 256 + 4 pad dwords -> conflict-free ds_load_b64
#define SQ_STRIDE 260

#if defined(__has_builtin)
#if __has_builtin(__builtin_amdgcn_tensor_load_to_lds) && __has_builtin(__builtin_amdgcn_s_wait_tensorcnt)
#define HAVE_TDM 1
#endif
#endif
#ifndef HAVE_TDM
#define HAVE_TDM 0
#endif

// ---- tiny precursor: q_eff[16][256], b_eff[16] into workspace -------------
__global__ void build_qeff_kernel(const float* __restrict__ Q,
                                  const float* __restrict__ Wk,
                                  const float* __restrict__ bk,
                                  float* __restrict__ qeff,
                                  float* __restrict__ beff) {
  const int i = threadIdx.x;            // 256 threads, one block
  for (int h = 0; h < H_; ++h) {
    float acc = 0.f;
#pragma unroll
    for (int d = 0; d < DK_; ++d)
      acc = fmaf(Q[h * DK_ + d], Wk[(size_t)(h * DK_ + d) * DIN_ + i], acc);
    qeff[h * DIN_ + i] = acc;
  }
  if (i < H_) {
    float acc = 0.f;
#pragma unroll
    for (int d = 0; d < DK_; ++d)
      acc = fmaf(Q[i * DK_ + d], bk[i * DK_ + d], acc);
    beff[i] = acc;
  }
}

// ---- main fused kernel: one block (256 thr / 8 waves) per batch -----------
__launch_bounds__(256)
__global__ void lmha_fused_kernel(const float* __restrict__ v,
                                  const unsigned char* __restrict__ pad_mask,
                                  const float* __restrict__ qeff,
                                  const float* __restrict__ beff,
                                  float* __restrict__ out,
                                  int B) {
  __shared__ float sv[SROW * SV_STRIDE];   // v tile, padded stride (66.5 KB)
  __shared__ float sq[H_ * SQ_STRIDE];     // q_eff, padded stride  (16.6 KB)
  __shared__ float ssc[H_ * 64];           // scores -> attn [head][s]
  __shared__ float sinv[H_];
  __shared__ float sb[H_];

  const int tid = threadIdx.x;
  const int b   = blockIdx.x;
  const float* vb = v + (size_t)b * (S_ * DIN_);

  // ---- stage v[b] (60x256 f32) into LDS with padded stride ----------------
#if HAVE_TDM
  if (tid < 32) {  // wave 0 issues one TDM descriptor (EXEC ignored by TDM)
    unsigned long long ga = (unsigned long long)(const void*)vb;
    unsigned lds_off      = (unsigned)(unsigned long long)(const void*)&sv[0];
    // D# group0: count=1, lds_addr, global_addr[56:0], type=2
    v4u g0 = { 1u,
               lds_off,
               (unsigned)(ga & 0xFFFFFFFFull),
               (unsigned)((ga >> 32) & 0x01FFFFFFull) | (2u << 30) };
    // D# group1: data_size=4B(2), pad_enable, pad_interval=256dw(7),
    //            pad_amount=4dw(3), tensor_dim0=256, tensor_dim1=60,
    //            tile_dim0=256, tile_dim1=60, tensor_dim0_stride=256
    v8i g1 = { (int)0x07D20000,          // mask=0|ds=2|pad_en|intv=7|amt=3
               (int)(256u << 16),        // tensor_dim0[15:0] @ bits[63:48]
               (int)(60u  << 16),        // tensor_dim1[15:0] @ bits[95:80]
               (int)(256u << 16),        // tile_dim0 @ bits[127:112]
               (int)60,                  // tile_dim1 @ bits[143:128]
               (int)256,                 // tensor_dim0_stride @ bits[191:160]
               0, 0 };
    v4i g2 = {0, 0, 0, 0};
    v4i g3 = {0, 0, 0, 0};
#if defined(__clang_major__) && (__clang_major__ >= 23)
    v8i gx = {0, 0, 0, 0, 0, 0, 0, 0};
    __builtin_amdgcn_tensor_load_to_lds(g0, g1, g2, g3, gx, 0);
#else
    __builtin_amdgcn_tensor_load_to_lds(g0, g1, g2, g3, 0);
#endif
    __builtin_amdgcn_s_wait_tensorcnt(0);
  }
#else
  for (int i = tid; i < S_ * (DIN_ / 4); i += 256) {   // 3840 float4
    int r = i >> 6, c = i & 63;
    ((float4*)&sv[r * SV_STRIDE])[c] = ((const float4*)vb)[i];
  }
#endif

  // q_eff -> LDS (padded stride), bias -> LDS, zero pad rows 60..63
  for (int i = tid; i < H_ * (DIN_ / 4); i += 256) {
    int r = i >> 6, c = i & 63;
    ((float4*)&sq[r * SQ_STRIDE])[c] = ((const float4*)qeff)[i];
  }
  if (tid < H_) sb[tid] = beff[tid];
  for (int i = tid; i < (SROW - S_) * SV_STRIDE; i += 256)
    sv[S_ * SV_STRIDE + i] = 0.0f;
  __syncthreads();

  // ---- scores[64,16] = sv[64,256] @ sq^T[256,16] via WMMA f32 16x16x4 -----
  const int wave = tid >> 5;
  const int lane = tid & 31;
  if (wave < 4) {
    const int lane_lo = lane & 15;       // A: M row ; B/C: N col (head)
    const int lane_hi = lane >> 4;       // K-half select / C row-half
    const int row = wave * 16 + lane_lo; // s position of this lane's A row
    const float* arow = &sv[row * SV_STRIDE];
    const float* brow = &sq[lane_lo * SQ_STRIDE];
    v8f c = {};
#pragma unroll
    for (int k0 = 0; k0 < DIN_; k0 += 4) {
      v2f a = *(const v2f*)(arow + k0 + 2 * lane_hi);  // A[M][k0+2lh..+1]
      v2f bfr = *(const v2f*)(brow + k0 + 2 * lane_hi);// B[k0+2lh..][N]
      c = __builtin_amdgcn_wmma_f32_16x16x4_f32(
              /*neg_a=*/false, a, /*neg_b=*/false, bfr,
              /*c_mod=*/(short)0, c, /*reuse_a=*/false, /*reuse_b=*/false);
    }
    // C layout: VGPR r -> M = r + 8*lane_hi (within tile), N = lane_lo
#pragma unroll
    for (int r = 0; r < 8; ++r) {
      int s = wave * 16 + r + 8 * lane_hi;
      ssc[lane_lo * 64 + s] = c[r];
    }
  }
  __syncthreads();

  // ---- bias + 1/sqrt(8) scale + pad mask ----------------------------------
  const float inv_temp = 0.35355339059327379f;
  for (int i = tid; i < H_ * S_; i += 256) {
    int h = i / S_, s = i - h * S_;
    float val = (ssc[h * 64 + s] + sb[h]) * inv_temp;
    if (pad_mask[(size_t)b * S_ + s]) val = -1.0e6f;
    ssc[h * 64 + s] = val;
  }
  __syncthreads();

  // ---- per-head softmax reduction (16 lanes, 60 elems each) ---------------
  if (tid < H_) {
    const int h = tid;
    float mx = -3.4e38f;
    for (int s = 0; s < S_; ++s) mx = fmaxf(mx, ssc[h * 64 + s]);
    float sum = 0.f;
    for (int s = 0; s < S_; ++s) {
      float e = __expf(ssc[h * 64 + s] - mx);
      ssc[h * 64 + s] = e;
      sum += e;
    }
    sinv[h] = 1.0f / sum;
  }
  __syncthreads();

  // ---- normalize, emit attn [H,B,S] ---------------------------------------
  float* attn_out = out + (size_t)B * DIN_;
  for (int i = tid; i < H_ * S_; i += 256) {
    int h = i / S_, s = i - h * S_;
    float a = ssc[h * 64 + s] * sinv[h];
    ssc[h * 64 + s] = a;
    attn_out[((size_t)h * B + b) * S_ + s] = a;
  }
  __syncthreads();

  // ---- out[b, h*16+d] = sum_s attn[h,s] * v[b,s,h*16+d] -------------------
  {
    const int h = tid >> 4, d = tid & 15;
    float acc = 0.f;
#pragma unroll 4
    for (int s = 0; s < S_; ++s)
      acc = fmaf(ssc[h * 64 + s], sv[s * SV_STRIDE + h * 16 + d], acc);
    out[(size_t)b * DIN_ + tid] = acc;
  }
}

// ---------------------------------------------------------------------------
extern "C" void kernel_launch(void* const* d_in, const int* in_sizes, int n_in,
                              void* d_out, int out_size, void* d_ws, size_t ws_size,
                              hipStream_t stream) {
  const float*         vptr = (const float*)d_in[0];
  const float*         Q    = (const float*)d_in[1];
  const float*         Wk   = (const float*)d_in[2];
  const float*         bk   = (const float*)d_in[3];
  const unsigned char* mask = (const unsigned char*)d_in[4];
  float* out  = (float*)d_out;
  float* qeff = (float*)d_ws;
  float* beff = qeff + H_ * DIN_;
  const int B = in_sizes[0] / (S_ * DIN_);

  build_qeff_kernel<<<1, 256, 0, stream>>>(Q, Wk, bk, qeff, beff);
  lmha_fused_kernel<<<B, 256, 0, stream>>>(vptr, mask, qeff, beff, out, B);
}